// WfcNN_44032004719009
// MI455X (gfx1250) — compile-verified
//
#include <hip/hip_runtime.h>

// CDNA5 / gfx1250, wave32. Full-fp32 WMMA path: V_WMMA_F32_16X16X4_F32.
typedef __attribute__((ext_vector_type(2))) float v2f;
typedef __attribute__((ext_vector_type(8))) float v8f;

__device__ __forceinline__ v8f wmma_f32_k4(v2f a, v2f b, v8f c) {
  // 8-arg form: (neg_a, A, neg_b, B, c_mod, C, reuse_a, reuse_b)
  return __builtin_amdgcn_wmma_f32_16x16x4_f32(false, a, false, b, (short)0, c,
                                               false, false);
}

__device__ __forceinline__ float fast_tanh(float x) {
  // tanh(x) = 1 - 2/(exp(2x)+1), built from native TRANS ops:
  // v_exp_f32 (exp2 with 2/ln2 folded into the scale) and v_rcp_f32.
  // TRANS ops co-execute with the WMMA pipe; avoids the IEEE div expansion.
  float e = __builtin_amdgcn_exp2f(x * 2.8853900817779268f);  // exp(2x)
  float r = __builtin_amdgcn_rcpf(e + 1.0f);
  return __builtin_fmaf(-2.0f, r, 1.0f);
}

// ---------------------------------------------------------------------------
// Pass 0b: rewrite HW1 into WMMA B-fragment pair layout:
//   HW1t pair (h, p, c) = { HW1[h][2p][c], HW1[h][2p+1][c] },  p<32, c<32.
// One aligned b64 load then yields a complete B-fragment VGPR pair.
// ---------------------------------------------------------------------------
__global__ void __launch_bounds__(256) k_transposeHW1(
    const float* __restrict__ HW1, v2f* __restrict__ HW1t) {
  const int q = blockIdx.x * 256 + threadIdx.x;  // pair index, 30*1024 total
  if (q >= 30 * 1024) return;
  const int h = q >> 10;
  const int p = (q >> 5) & 31;
  const int c = q & 31;
  v2f w;
  w.x = HW1[h * 2048 + (2 * p) * 32 + c];
  w.y = HW1[h * 2048 + (2 * p + 1) * 32 + c];
  HW1t[q] = w;
}

// ---------------------------------------------------------------------------
// Pass 1: shared MLP 2->128(tanh)->64(tanh), emit h2[B,64], hid[B], histogram.
// One wave per 16-sample tile; layer2 done as [16,128]x[128,64] with
// 32 K-steps x 4 N-tiles of V_WMMA_F32_16X16X4_F32. Layer1 (K=2) is fused
// into per-lane A-fragment generation. W2s is staged in LDS pre-swizzled
// into B-fragment pairs so each fragment is a single aligned b64 DS load.
// ---------------------------------------------------------------------------
__global__ void __launch_bounds__(256) k_shared_mlp(
    const float* __restrict__ x, const int* __restrict__ qn,
    const int* __restrict__ ql, const int* __restrict__ qm,
    const float* __restrict__ W1, const float* __restrict__ b1,
    const float* __restrict__ W2, const float* __restrict__ b2,
    float* __restrict__ h2, int* __restrict__ hid, int* __restrict__ ghist,
    int B) {
  __shared__ float sW1[2 * 128];
  __shared__ float sb1[128];
  __shared__ v2f sW2p[64 * 64];   // pair (p,c): {W2[2p][c], W2[2p+1][c]}
  __shared__ float sb2[64];
  __shared__ int shist[32];

  const int t = threadIdx.x;
  if (t < 32) shist[t] = 0;
  sW1[t] = W1[t];                       // 256 threads == 256 elems
  if (t < 128) sb1[t] = b1[t];
  for (int q = t; q < 64 * 64; q += 256) {
    const int p = q >> 6, c = q & 63;
    v2f w;
    w.x = W2[(2 * p) * 64 + c];
    w.y = W2[(2 * p + 1) * 64 + c];
    sW2p[q] = w;
  }
  if (t < 64) sb2[t] = b2[t];
  __syncthreads();

  const int lane = t & 31;
  const int half = lane >> 4;           // 0: K pair {0,1}; 1: K pair {2,3}
  const int r = lane & 15;
  const int wave = (blockIdx.x * 256 + t) >> 5;
  const int sampleBase = wave * 16;

  if (sampleBase < B) {                 // wave-uniform -> EXEC all-ones inside
    const int sA = sampleBase + r;      // this lane's A-matrix row (sample)
    const float x0 = x[sA * 2 + 0];
    const float x1 = x[sA * 2 + 1];

    v8f acc0 = {}, acc1 = {}, acc2 = {}, acc3 = {};
#pragma unroll 4
    for (int k = 0; k < 32; ++k) {
      const int j0 = k * 4 + half * 2;  // h1 feature pair this lane provides
      const int p = 2 * k + half;       // pair row in sW2p
      v2f A;
      A.x = fast_tanh(fmaf(x0, sW1[j0],     fmaf(x1, sW1[128 + j0],     sb1[j0])));
      A.y = fast_tanh(fmaf(x0, sW1[j0 + 1], fmaf(x1, sW1[128 + j0 + 1], sb1[j0 + 1])));
      const v2f B0 = sW2p[p * 64 + 0 + r];
      const v2f B1 = sW2p[p * 64 + 16 + r];
      const v2f B2 = sW2p[p * 64 + 32 + r];
      const v2f B3 = sW2p[p * 64 + 48 + r];
      acc0 = wmma_f32_k4(A, B0, acc0);
      acc1 = wmma_f32_k4(A, B1, acc1);
      acc2 = wmma_f32_k4(A, B2, acc2);
      acc3 = wmma_f32_k4(A, B3, acc3);
    }

    v8f accs[4] = {acc0, acc1, acc2, acc3};
#pragma unroll
    for (int n = 0; n < 4; ++n) {
      const float bb = sb2[n * 16 + r];
      v8f a = accs[n];
#pragma unroll
      for (int i = 0; i < 8; ++i) {
        const int row = i + half * 8;   // C layout: VGPR i, lane half -> M
        h2[(size_t)(sampleBase + row) * 64 + n * 16 + r] =
            fast_tanh(a[i] + bb);
      }
    }

    if (half == 0) {                    // lanes 0..15: head id + histogram
      const int s = sampleBase + r;
      const int nn = qn[s], ll = ql[s], mm = qm[s];
      const int hd = (nn - 1) * nn * (2 * nn - 1) / 6 + ll * ll + (mm + ll);
      hid[s] = hd;
      atomicAdd(&shist[hd], 1);
    }
  }
  __syncthreads();
  if (t < 30) {
    const int c = shist[t];
    if (c) atomicAdd(&ghist[t], c);
  }
}

// ---------------------------------------------------------------------------
// Pass 2: tiny serial scan over 30 heads -> offsets, cursors, tile offsets.
// ---------------------------------------------------------------------------
__global__ void k_scan(const int* __restrict__ ghist, int* __restrict__ offsets,
                       int* __restrict__ cursor, int* __restrict__ tileOff) {
  if (threadIdx.x == 0 && blockIdx.x == 0) {
    int acc = 0, tacc = 0;
    for (int h = 0; h < 30; ++h) {
      const int c = ghist[h];
      offsets[h] = acc;
      cursor[h] = acc;
      tileOff[h] = tacc;
      acc += c;
      tacc += (c + 15) >> 4;
    }
    offsets[30] = acc;
    tileOff[30] = tacc;
  }
}

// ---------------------------------------------------------------------------
// Pass 3: head-grouped permutation. LDS rank per block + one global atomic
// per (head,block). Output is permutation-invariant per sample, so atomic
// ordering nondeterminism does not change final results.
// ---------------------------------------------------------------------------
__global__ void __launch_bounds__(256) k_scatter(const int* __restrict__ hid,
                                                 int* __restrict__ cursor,
                                                 int* __restrict__ perm,
                                                 int B) {
  __shared__ int lcount[32];
  __shared__ int lbase[32];
  const int t = threadIdx.x;
  if (t < 32) lcount[t] = 0;
  __syncthreads();
  const int s = blockIdx.x * 256 + t;
  int hd = 0, lrank = 0;
  const bool valid = s < B;
  if (valid) {
    hd = hid[s];
    lrank = atomicAdd(&lcount[hd], 1);
  }
  __syncthreads();
  if (t < 30) {
    const int c = lcount[t];
    lbase[t] = c ? atomicAdd(&cursor[t], c) : 0;
  }
  __syncthreads();
  if (valid) perm[lbase[hd] + lrank] = s;
}

// ---------------------------------------------------------------------------
// Pass 4: per-head GEMM. One wave per 16-sample tile of one head:
// [16,64] x HW1[h](64x32) via 16 K-steps x 2 N-tiles of f32 WMMA (B-frags
// come pre-paired from HW1t as single b64 loads), tanh, then the 32->2
// layer per-lane (16 samples x 2 outputs == 32 lanes) after an in-wave LDS
// transpose (DS ops are in-order per wave; explicit s_wait_dscnt).
// Tail tiles clamp row index -> duplicate lanes write identical values.
// ---------------------------------------------------------------------------
__global__ void __launch_bounds__(256) k_heads(
    const float* __restrict__ h2, const int* __restrict__ perm,
    const int* __restrict__ offsets, const int* __restrict__ tileOff,
    const v2f* __restrict__ HW1t, const float* __restrict__ Hb1,
    const float* __restrict__ HW2, const float* __restrict__ Hb2,
    float* __restrict__ out) {
  __shared__ float zbuf[8][16][32];
  const int t = threadIdx.x;
  const int wv = t >> 5;
  const int lane = t & 31;
  const int half = lane >> 4;
  const int r = lane & 15;

  const int tile = blockIdx.x * 8 + wv;
  const int total = tileOff[30];
  if (tile >= total) return;            // wave-uniform exit, no block barriers

  int h = 0;
#pragma unroll
  for (int i = 1; i < 30; ++i)
    if (tile >= tileOff[i]) h = i;
  const int ti = tile - tileOff[h];
  const int rowBase = offsets[h] + ti * 16;
  int validRows = (offsets[h + 1] - offsets[h]) - ti * 16;
  if (validRows > 16) validRows = 16;
  const int rr = (r < validRows) ? r : (validRows - 1);
  const int s = perm[rowBase + rr];

  const v2f* __restrict__ Arow = (const v2f*)(h2 + (size_t)s * 64);
  const v2f* __restrict__ Whp = HW1t + h * 1024;

  v8f acc0 = {}, acc1 = {};
#pragma unroll
  for (int k = 0; k < 16; ++k) {
    const int p = 2 * k + half;         // pair row: h2 features {2p, 2p+1}
    const v2f A  = Arow[p];
    const v2f B0 = Whp[p * 32 + r];
    const v2f B1 = Whp[p * 32 + 16 + r];
    acc0 = wmma_f32_k4(A, B0, acc0);
    acc1 = wmma_f32_k4(A, B1, acc1);
  }

  const float bb0 = Hb1[h * 32 + r];
  const float bb1 = Hb1[h * 32 + 16 + r];
#pragma unroll
  for (int i = 0; i < 8; ++i) {
    const int row = i + half * 8;
    zbuf[wv][row][r]      = fast_tanh(acc0[i] + bb0);
    zbuf[wv][row][16 + r] = fast_tanh(acc1[i] + bb1);
  }
  asm volatile("s_wait_dscnt 0" ::: "memory");  // in-wave LDS transpose fence

  // 32->2 output layer: lane = (sample r, output half)
  const int o = half;
  float a = Hb2[h * 2 + o];
  const float* __restrict__ W2h = HW2 + h * 64;
#pragma unroll
  for (int i = 0; i < 32; ++i) a = fmaf(zbuf[wv][r][i], W2h[i * 2 + o], a);
  out[s * 2 + o] = a;
}

__global__ void k_init(int* __restrict__ ghist) {
  if (threadIdx.x < 32) ghist[threadIdx.x] = 0;
}

extern "C" void kernel_launch(void* const* d_in, const int* in_sizes, int n_in,
                              void* d_out, int out_size, void* d_ws,
                              size_t ws_size, hipStream_t stream) {
  const float* x   = (const float*)d_in[0];
  const int*   qn  = (const int*)d_in[1];
  const int*   ql  = (const int*)d_in[2];
  const int*   qm  = (const int*)d_in[3];
  const float* W1s = (const float*)d_in[4];
  const float* b1s = (const float*)d_in[5];
  const float* W2s = (const float*)d_in[6];
  const float* b2s = (const float*)d_in[7];
  const float* HW1 = (const float*)d_in[8];
  const float* Hb1 = (const float*)d_in[9];
  const float* HW2 = (const float*)d_in[10];
  const float* Hb2 = (const float*)d_in[11];
  float* out = (float*)d_out;

  const int B = in_sizes[1];  // n[] has B elements

  char* ws = (char*)d_ws;
  float* h2   = (float*)ws;                          // B*64 fp32
  int*   hid  = (int*)(ws + (size_t)B * 64 * 4);     // B ints
  int*   perm = hid + B;                             // B ints
  int*   meta = perm + B;
  int* ghist   = meta;       // 30
  int* offsets = meta + 32;  // 31
  int* cursor  = meta + 64;  // 30
  int* tileOff = meta + 96;  // 31
  v2f* HW1t = (v2f*)(meta + 128);  // 30*1024 pairs (8B-aligned offset)

  k_init<<<1, 32, 0, stream>>>(ghist);
  k_transposeHW1<<<(30 * 1024 + 255) / 256, 256, 0, stream>>>(HW1, HW1t);

  const int waves1 = (B + 15) / 16;
  const int blocks1 = (waves1 + 7) / 8;
  k_shared_mlp<<<blocks1, 256, 0, stream>>>(x, qn, ql, qm, W1s, b1s, W2s, b2s,
                                            h2, hid, ghist, B);

  k_scan<<<1, 32, 0, stream>>>(ghist, offsets, cursor, tileOff);

  k_scatter<<<(B + 255) / 256, 256, 0, stream>>>(hid, cursor, perm, B);

  const int maxTiles = B / 16 + 30;
  k_heads<<<(maxTiles + 7) / 8, 256, 0, stream>>>(h2, perm, offsets, tileOff,
                                                  HW1t, Hb1, HW2, Hb2, out);
}